// ARX_69123203662661
// MI455X (gfx1250) — compile-verified
//
#include <hip/hip_runtime.h>

// ARX recurrence, blocked into 16-step chunks so each chunk is a 16x16x16 f32
// GEMM done with 4x V_WMMA_F32_16X16X4_F32 per 16-row batch tile (wave32).
//
// Math: window W (16 rows x 16 cols). 16 consecutive preds: P = W * M^T + const (x) v,
// and the new window is exactly P. M, v precomputed from w_ar in a setup kernel.

typedef __attribute__((ext_vector_type(2))) float v2f;
typedef __attribute__((ext_vector_type(8))) float v8f;

#define AR   16
#define UD   15
#define SEQ  1024
#define WAVES_PER_BLOCK 8
#define BLOCK_THREADS   256

// ---------------------------------------------------------------------------
// Setup: compute M[16][16] and v[16] such that
//   pred_k = sum_i M[k][i] * s_i + v[k] * const
// over the 16-entry starting window s. Stored in ws: M row-major at [0..255],
// v at [256..271].
// ---------------------------------------------------------------------------
__global__ void arx_setup_kernel(const float* __restrict__ w,
                                 float* __restrict__ ws) {
  if (threadIdx.x != 0 || blockIdx.x != 0) return;
  float wa[AR];
  for (int i = 0; i < AR; ++i) wa[i] = w[i];

  float M[AR][AR];
  float v[AR];
  for (int k = 0; k < AR; ++k) {
    float row[AR];
    for (int i = 0; i < AR; ++i) row[i] = 0.0f;
    float vk = 1.0f;
    for (int j = 0; j < AR; ++j) {
      if (j < AR - k) {
        // window_k[j] = s_{k+j}
        row[k + j] += wa[j];
      } else {
        // window_k[j] = pred_{j-(AR-k)}
        int p = j - (AR - k);
        for (int i = 0; i < AR; ++i) row[i] += wa[j] * M[p][i];
        vk += wa[j] * v[p];
      }
    }
    for (int i = 0; i < AR; ++i) M[k][i] = row[i];
    v[k] = vk;
  }
  for (int k = 0; k < AR; ++k)
    for (int i = 0; i < AR; ++i)
      ws[k * AR + i] = M[k][i];
  for (int k = 0; k < AR; ++k) ws[256 + k] = v[k];
}

// ---------------------------------------------------------------------------
// Main kernel: one wave per 16 batch rows. 63 block steps, each = 4 WMMAs.
//
// WMMA mapping (D = A x B + C, 16x16x4 f32, chained over 4 K-chunks):
//   A[m][kk]  = W[batch row m][window idx kk]      (A layout: lane holds row lane%16,
//                                                   VGPR j of chunk c -> kk = 4c + j + 2*(lane/16))
//   B[kk][n]  = M[n][kk]                           (B layout mirrored: lane holds col lane%16)
//   C[m][n]   = const[m] * v[n]
// D layout (C/D 16x16 f32): VGPR r, lane L -> D[r + 8*(L/16)][L%16].
// ---------------------------------------------------------------------------
__global__ __launch_bounds__(BLOCK_THREADS) void arx_main_kernel(
    const float* __restrict__ y,    // [B, 16]
    const float* __restrict__ u,    // [B, 15]
    const float* __restrict__ w,    // [32] = w_ar(16) | w_u(15) | w_b
    const float* __restrict__ ws,   // M(256) | v(16)
    float* __restrict__ out)        // [B, 1024]
{
  __shared__ float s_tile[WAVES_PER_BLOCK][AR][AR];
  __shared__ float s_const[WAVES_PER_BLOCK][AR];

  const int lane = threadIdx.x & 31;
  const int wv   = threadIdx.x >> 5;
  const int lo   = lane & 15;     // 0..15
  const int hi   = lane >> 4;     // 0 or 1
  const size_t b0 = ((size_t)blockIdx.x * WAVES_PER_BLOCK + wv) * AR;

  // ---- per-row constant: const[b] = u[b] . w_u + w_b ----------------------
  {
    float acc = w[AR + UD];                       // w_b
    const float* up = u + (b0 + (size_t)lo) * UD;
#pragma unroll
    for (int i = 0; i < UD; ++i) acc += up[i] * w[AR + i];
    if (hi == 0) s_const[wv][lo] = acc;           // lanes 0..15 publish 16 rows
  }
  // wave-local LDS ordering: writes above visible to reads below
  asm volatile("s_wait_dscnt 0" ::: "memory");

  // ---- C init: cinit[r] = const[r + 8*hi] * v[lo] -------------------------
  const float vlane = ws[256 + lo];
  float cinit[8];
#pragma unroll
  for (int r = 0; r < 8; ++r) cinit[r] = s_const[wv][r + 8 * hi] * vlane;

  // ---- fixed B operand: B[kk][n] = M[n][kk], n = lo, kk = 4c + j + 2*hi ---
  v2f Bm[4];
#pragma unroll
  for (int c = 0; c < 4; ++c)
    Bm[c] = *(const v2f*)(ws + lo * AR + 4 * c + 2 * hi);

  // ---- initial A operand straight from y (A layout) -----------------------
  v2f Am[4];
#pragma unroll
  for (int c = 0; c < 4; ++c)
    Am[c] = *(const v2f*)(y + (b0 + (size_t)lo) * AR + 4 * c + 2 * hi);

  // ---- out[:, 0:16] = y ---------------------------------------------------
#pragma unroll
  for (int r = 0; r < 8; ++r) {
    const size_t row = b0 + r + 8 * hi;
    out[row * SEQ + lo] = y[row * AR + lo];
  }

  // ---- 63 block steps of 16 predictions each ------------------------------
  const int NBLK = (SEQ - AR) / AR;   // 63
  for (int it = 0; it < NBLK; ++it) {
    v8f acc;
#pragma unroll
    for (int r = 0; r < 8; ++r) acc[r] = cinit[r];

#pragma unroll
    for (int c = 0; c < 4; ++c)
      acc = __builtin_amdgcn_wmma_f32_16x16x4_f32(
          /*neg_a=*/false, Am[c], /*neg_b=*/false, Bm[c],
          /*c_mod=*/(short)0, acc, /*reuse_a=*/false, /*reuse_b=*/false);

    // store 16x16 prediction tile: cols [16+16*it .. +15]
    const int col0 = AR + it * AR;
#pragma unroll
    for (int r = 0; r < 8; ++r)
      out[(b0 + r + 8 * hi) * SEQ + col0 + lo] = acc[r];

    // D-layout -> A-layout via per-wave LDS round trip (W := P)
#pragma unroll
    for (int r = 0; r < 8; ++r)
      s_tile[wv][r + 8 * hi][lo] = acc[r];
    asm volatile("s_wait_dscnt 0" ::: "memory");   // store -> load ordering
#pragma unroll
    for (int c = 0; c < 4; ++c)
      Am[c] = *(const v2f*)(&s_tile[wv][lo][4 * c + 2 * hi]);
    asm volatile("" ::: "memory");                 // keep next-iter stores after loads
  }
}

// ---------------------------------------------------------------------------
extern "C" void kernel_launch(void* const* d_in, const int* in_sizes, int n_in,
                              void* d_out, int out_size, void* d_ws, size_t ws_size,
                              hipStream_t stream) {
  const float* y = (const float*)d_in[0];   // [B,16]
  const float* u = (const float*)d_in[1];   // [B,15]
  const float* w = (const float*)d_in[2];   // [32]
  float* out = (float*)d_out;               // [B,1024]
  float* ws  = (float*)d_ws;                // >= 272 floats

  const int B = in_sizes[0] / AR;           // 65536
  arx_setup_kernel<<<1, 32, 0, stream>>>(w, ws);

  const int rows_per_block = WAVES_PER_BLOCK * AR;  // 128
  arx_main_kernel<<<B / rows_per_block, BLOCK_THREADS, 0, stream>>>(y, u, w, ws, out);
}